// vsa_layer_36867999268878
// MI455X (gfx1250) — compile-verified
//
#include <hip/hip_runtime.h>

#define B_ 2
#define T_ 12
#define N_ 1024
#define D_ 64
#define H_ 4
#define HD_ 16
#define EPS_ 1e-5f
#define NEG_BIG -3.0e38f

#define CHUNK_ 256                 // key columns per chunk
#define NCHUNK_ (N_ / CHUNK_)      // 4
#define NT_ (CHUNK_ / 16)          // 16 tiles per chunk
#define SSTRIDE_ 260               // 256 + 4 pad; 260 % 64 == 4 -> conflict-free

typedef __attribute__((ext_vector_type(2))) float v2f;
typedef __attribute__((ext_vector_type(8))) float v8f;

__device__ __forceinline__ v8f wmma_f32_k4(v2f a, v2f b, v8f c) {
    // D = A(16x4,f32) * B(4x16,f32) + C(16x16,f32)
    return __builtin_amdgcn_wmma_f32_16x16x4_f32(
        /*neg_a=*/false, a, /*neg_b=*/false, b,
        /*c_mod=*/(short)0, c, /*reuse_a=*/false, /*reuse_b=*/false);
}

// ---------------------------------------------------------------------------
// Kernel 1: Q/K/V = x @ W^T + b   (one wave per 16-row tile, blockIdx.y picks W)
// Q is pre-scaled by 1/sqrt(HD) = 0.25 so the attention kernel skips it.
// ---------------------------------------------------------------------------
__global__ __launch_bounds__(32) void qkv_kernel(
    const float* __restrict__ x,
    const float* __restrict__ Wq, const float* __restrict__ bq,
    const float* __restrict__ Wk, const float* __restrict__ bk,
    const float* __restrict__ Wv, const float* __restrict__ bv,
    float* __restrict__ Q, float* __restrict__ Ko, float* __restrict__ V)
{
    __shared__ float xs[16 * 68];              // 16x64 x-tile, padded stride 68
    const int lane = threadIdx.x & 31;
    const int half = lane >> 4;                // 0 or 1
    const int l16  = lane & 15;
    const int which = blockIdx.y;              // 0=Q 1=K 2=V
    const float* W    = (which == 0) ? Wq : (which == 1) ? Wk : Wv;
    const float* bias = (which == 0) ? bq : (which == 1) ? bk : bv;
    float*       out  = (which == 0) ? Q  : (which == 1) ? Ko : V;
    const float  osc  = (which == 0) ? 0.25f : 1.0f;   // fold 1/sqrt(HD) into Q

    const size_t row0 = (size_t)blockIdx.x * 16;
    const float* xt = x + row0 * D_;

    // coalesced load of 16x64 floats into padded LDS
    #pragma unroll
    for (int i = 0; i < 8; ++i) {
        const int idx = (i * 32 + lane) * 4;   // float index, 16B aligned
        const float4 val = *(const float4*)(xt + idx);
        const int row = idx >> 6, col = idx & 63;
        float* dst = &xs[row * 68 + col];
        dst[0] = val.x; dst[1] = val.y; dst[2] = val.z; dst[3] = val.w;
    }
    __syncthreads();

    // A fragments: a[kk][i] = x[row0 + l16][kk*4 + i + 2*half]
    v2f afrag[16];
    #pragma unroll
    for (int kk = 0; kk < 16; ++kk) {
        const float* p = &xs[l16 * 68 + kk * 4 + 2 * half];
        v2f a; a.x = p[0]; a.y = p[1];
        afrag[kk] = a;
    }

    v8f acc[4];
    #pragma unroll
    for (int c = 0; c < 4; ++c) acc[c] = {};

    // b[i] = B[k = kk*4+i+2*half][col = c*16+l16] = W[(c*16+l16)*64 + k]  (W^T)
    #pragma unroll
    for (int kk = 0; kk < 16; ++kk) {
        #pragma unroll
        for (int c = 0; c < 4; ++c) {
            const float* wp = &W[(c * 16 + l16) * D_ + kk * 4 + 2 * half];
            v2f b; b.x = wp[0]; b.y = wp[1];
            acc[c] = wmma_f32_k4(afrag[kk], b, acc[c]);
        }
    }

    #pragma unroll
    for (int c = 0; c < 4; ++c) {
        const float bb = bias[c * 16 + l16];
        #pragma unroll
        for (int r = 0; r < 8; ++r) {
            out[(row0 + r + 8 * half) * D_ + c * 16 + l16] = (acc[c][r] + bb) * osc;
        }
    }
}

// ---------------------------------------------------------------------------
// Kernel 2: flash attention per (bt, head, 16-query-row tile)
// scores = (Qs.K^T) * adj   (Qs already carries the 0.25 scale)
// Four 256-column chunks; per chunk: pass1 scores->LDS + running max,
// one rescale, pass2 exp + PV WMMA reading A-fragments straight from LDS.
// ~16.6KB LDS/wave -> ~19 resident workgroups per WGP for latency hiding.
// ---------------------------------------------------------------------------
__global__ __launch_bounds__(32) void attn_kernel(
    const float* __restrict__ Qs, const float* __restrict__ K,
    const float* __restrict__ V, const float* __restrict__ adj,
    float* __restrict__ Hout)
{
    __shared__ float s_lds[16 * SSTRIDE_];     // 16 rows x CHUNK_ scores
    __shared__ float mx_bc[16];                // per-row max broadcast
    __shared__ float sc_bc[16];                // per-row rescale broadcast
    __shared__ float rs_bc[16];                // per-row denom broadcast
    const int lane = threadIdx.x & 31;
    const int half = lane >> 4;
    const int l16  = lane & 15;
    const int n0 = blockIdx.x * 16;            // query row tile
    const int h  = blockIdx.y;                 // head
    const int bt = blockIdx.z;                 // fused batch*time
    const size_t base  = (size_t)bt * N_ * D_;
    const size_t abase = (size_t)bt * N_ * N_;

    // Q fragments: a[kk][i] = Qs[n0+l16][h*16 + kk*4 + i + 2*half]
    v2f aq[4];
    #pragma unroll
    for (int kk = 0; kk < 4; ++kk) {
        const float* p = &Qs[base + (size_t)(n0 + l16) * D_ + h * HD_ + kk * 4 + 2 * half];
        v2f a; a.x = p[0]; a.y = p[1];
        aq[kk] = a;
    }

    float mrow[8];
    #pragma unroll
    for (int r = 0; r < 8; ++r) mrow[r] = NEG_BIG;
    float lsum = 0.0f;                         // A-layout partial denom (row l16)
    v8f acco = {};

    for (int chunk = 0; chunk < NCHUNK_; ++chunk) {
        // ---------------- pass 1: S = Qs.K^T * adj -> LDS, track max ----------
        float mxl[8];
        #pragma unroll
        for (int r = 0; r < 8; ++r) mxl[r] = NEG_BIG;

        for (int t = 0; t < NT_; ++t) {
            const int m0 = chunk * CHUNK_ + t * 16;
            v8f accs = {};
            #pragma unroll
            for (int kk = 0; kk < 4; ++kk) {
                const float* p = &K[base + (size_t)(m0 + l16) * D_ + h * HD_ + kk * 4 + 2 * half];
                v2f b; b.x = p[0]; b.y = p[1];
                accs = wmma_f32_k4(aq[kk], b, accs);
            }
            #pragma unroll
            for (int r = 0; r < 8; ++r) {
                const int n = n0 + r + 8 * half;
                const float a = __builtin_nontemporal_load(
                    &adj[abase + (size_t)n * N_ + m0 + l16]);   // read-once stream
                const float s = accs[r] * a;
                mxl[r] = fmaxf(mxl[r], s);
                s_lds[(r + 8 * half) * SSTRIDE_ + t * 16 + l16] = s;
            }
        }

        // ---------------- chunk max reduce + rescale (C-layout) ---------------
        #pragma unroll
        for (int r = 0; r < 8; ++r) {
            float mx = mxl[r];
            mx = fmaxf(mx, __shfl_xor(mx, 1));
            mx = fmaxf(mx, __shfl_xor(mx, 2));
            mx = fmaxf(mx, __shfl_xor(mx, 4));
            mx = fmaxf(mx, __shfl_xor(mx, 8));
            const float mnew = fmaxf(mrow[r], mx);
            const float sc   = __expf(mrow[r] - mnew);
            acco[r] *= sc;
            mrow[r] = mnew;
            if (l16 == r) {                    // one writer per row
                mx_bc[r + 8 * half] = mnew;
                sc_bc[r + 8 * half] = sc;
            }
        }
        __syncthreads();

        // ---------------- pass 2: exp + PV WMMA (A-layout from LDS) -----------
        const float mxr = mx_bc[l16];          // this lane's A-row max
        float csum = 0.0f;
        for (int t = 0; t < NT_; ++t) {
            const int m0 = chunk * CHUNK_ + t * 16;
            #pragma unroll
            for (int kk = 0; kk < 4; ++kk) {
                const v2f sv = *(const v2f*)&s_lds[l16 * SSTRIDE_ + t * 16 + kk * 4 + 2 * half];
                const float p0 = __expf(sv.x - mxr);
                const float p1 = __expf(sv.y - mxr);
                csum += p0 + p1;
                v2f ap; ap.x = p0; ap.y = p1;
                const float* vp = &V[base + (size_t)(m0 + kk * 4 + 2 * half) * D_ + h * HD_ + l16];
                v2f bv; bv.x = vp[0]; bv.y = vp[D_];
                acco = wmma_f32_k4(ap, bv, acco);
            }
        }
        lsum = lsum * sc_bc[l16] + csum;
        __syncthreads();                       // before next chunk reuses s_lds/bc
    }

    // final denom: combine the two lane-half partials (k%4 < 2 / k%4 >= 2) of row l16, broadcast
    const float rs = lsum + __shfl_xor(lsum, 16);
    if (half == 0) rs_bc[l16] = rs;
    __syncthreads();

    #pragma unroll
    for (int r = 0; r < 8; ++r) {
        const float inv = 1.0f / rs_bc[r + 8 * half];
        Hout[base + (size_t)(n0 + r + 8 * half) * D_ + h * HD_ + l16] = acco[r] * inv;
    }
}

// ---------------------------------------------------------------------------
// Kernel 3: y = h + x ; LayerNorm over D=64 ; one wave per row (2 elems/lane)
// ---------------------------------------------------------------------------
__global__ __launch_bounds__(256) void ln_kernel(
    const float* __restrict__ Hout, const float* __restrict__ x,
    const float* __restrict__ gamma, const float* __restrict__ beta,
    float* __restrict__ out)
{
    const int lane = threadIdx.x & 31;
    const int wid  = threadIdx.x >> 5;
    const size_t row = (size_t)blockIdx.x * 8 + wid;
    const float* hp = Hout + row * D_;
    const float* xp = x    + row * D_;
    const float y0 = hp[lane]      + xp[lane];
    const float y1 = hp[lane + 32] + xp[lane + 32];
    float s1 = y0 + y1;
    float s2 = y0 * y0 + y1 * y1;
    #pragma unroll
    for (int m = 1; m < 32; m <<= 1) {
        s1 += __shfl_xor(s1, m);
        s2 += __shfl_xor(s2, m);
    }
    const float mean = s1 * (1.0f / 64.0f);
    const float var  = s2 * (1.0f / 64.0f) - mean * mean;
    const float inv  = rsqrtf(var + EPS_);
    float* op = out + row * D_;
    op[lane]      = (y0 - mean) * inv * gamma[lane]      + beta[lane];
    op[lane + 32] = (y1 - mean) * inv * gamma[lane + 32] + beta[lane + 32];
}

extern "C" void kernel_launch(void* const* d_in, const int* in_sizes, int n_in,
                              void* d_out, int out_size, void* d_ws, size_t ws_size,
                              hipStream_t stream) {
    (void)in_sizes; (void)n_in; (void)out_size; (void)ws_size;
    const float* x     = (const float*)d_in[0];
    const float* adj   = (const float*)d_in[1];
    /* d_in[2] = s_adj (unused by reference) */
    const float* Wq    = (const float*)d_in[3];
    const float* bq    = (const float*)d_in[4];
    const float* Wk    = (const float*)d_in[5];
    const float* bk    = (const float*)d_in[6];
    const float* Wv    = (const float*)d_in[7];
    const float* bv    = (const float*)d_in[8];
    const float* gamma = (const float*)d_in[9];
    const float* beta  = (const float*)d_in[10];
    float* out = (float*)d_out;

    float* ws = (float*)d_ws;
    const size_t S = (size_t)B_ * T_ * N_ * D_;   // 1,572,864 floats
    float* Q  = ws;                                // pre-scaled by 0.25
    float* K  = ws + S;
    float* V  = ws + 2 * S;
    float* Hh = ws + 3 * S;

    dim3 g1(B_ * T_ * N_ / 16, 3, 1);
    qkv_kernel<<<g1, 32, 0, stream>>>(x, Wq, bq, Wk, bk, Wv, bv, Q, K, V);

    dim3 g2(N_ / 16, H_, B_ * T_);
    attn_kernel<<<g2, 32, 0, stream>>>(Q, K, V, adj, Hh);

    ln_kernel<<<(B_ * T_ * N_) / 8, 256, 0, stream>>>(Hh, x, gamma, beta, out);
}